// Hessian_6579889898139
// MI455X (gfx1250) — compile-verified
//
#include <hip/hip_runtime.h>

#ifndef __has_builtin
#define __has_builtin(x) 0
#endif

// ---- CDNA5 async global->LDS path (gfx1250), guarded so either toolchain compiles ----
#if __has_builtin(__builtin_amdgcn_global_load_async_to_lds_b32)
#define HAVE_ASYNC_LDS 1
typedef __attribute__((address_space(1))) int async_gptr_t;   // global
typedef __attribute__((address_space(3))) int async_lptr_t;   // LDS
// gfx12 CPol: TH in bits [2:0]; TH_NT = 1 (non-temporal: input is read-once,
// keep the 192MB L2 free for the 192MB output that kernel C re-reads)
#define ASYNC_CPOL_NT 1
#else
#define HAVE_ASYNC_LDS 0
#endif

#if __has_builtin(__builtin_amdgcn_s_wait_asynccnt)
#define WAIT_ASYNCCNT0() __builtin_amdgcn_s_wait_asynccnt(0)
#else
#define WAIT_ASYNCCNT0() asm volatile("s_wait_asynccnt 0" ::: "memory")
#endif

#define IMG_H 1024
#define IMG_W 1024
#define TILE 64
#define IW 68          // TILE + 4 (2-pixel halo each side)
#define SW 66          // TILE + 2 (1-pixel score halo for NMS)
#define NPLANES 48
#define PLANE_ELEMS (IMG_H * IMG_W)

typedef float v4f __attribute__((ext_vector_type(4)));

__global__ __launch_bounds__(256) void hessian_nms_kernel(const float* __restrict__ in,
                                                          float* __restrict__ out,
                                                          unsigned int* __restrict__ pmax) {
    __shared__ float s_in[IW * IW];   // 18.5 KB
    __shared__ float s_sc[SW * SW];   // 17.4 KB
    __shared__ float s_red[256];

    const int plane = blockIdx.z;
    const int bx = blockIdx.x * TILE;
    const int by = blockIdx.y * TILE;
    const int tid = threadIdx.x;
    const float* __restrict__ pin = in + (size_t)plane * PLANE_ELEMS;

    // ---- Stage (TILE+4)^2 input region into LDS with edge-replicate clamp ----
    // Async DMA (global -> LDS, no VGPR round-trip), non-temporal: input is
    // touched exactly once per tile, so don't let it evict output lines in L2.
    for (int idx = tid; idx < IW * IW; idx += 256) {
        int ly = idx / IW;
        int lx = idx - ly * IW;
        int gy = by + ly - 2; gy = min(max(gy, 0), IMG_H - 1);
        int gx = bx + lx - 2; gx = min(max(gx, 0), IMG_W - 1);
#if HAVE_ASYNC_LDS
        __builtin_amdgcn_global_load_async_to_lds_b32(
            (async_gptr_t*)(pin + gy * IMG_W + gx),
            (async_lptr_t*)&s_in[idx],
            /*offset=*/0, /*cpol=*/ASYNC_CPOL_NT);
#else
        s_in[idx] = __builtin_nontemporal_load(pin + gy * IMG_W + gx);
#endif
    }
#if HAVE_ASYNC_LDS
    WAIT_ASYNCCNT0();
#endif
    __syncthreads();

    // ---- Hessian det scores on (TILE+2)^2 region (0 outside the plane) ----
    for (int idx = tid; idx < SW * SW; idx += 256) {
        int sy = idx / SW;
        int sx = idx - sy * SW;
        int gy = by + sy - 1;
        int gx = bx + sx - 1;
        float v = 0.0f;
        if ((unsigned)gy < IMG_H && (unsigned)gx < IMG_W) {
            const float* c = &s_in[(sy + 1) * IW + (sx + 1)];
            float a00 = c[-IW - 1], a01 = c[-IW], a02 = c[-IW + 1];
            float a10 = c[-1],      a11 = c[0],   a12 = c[1];
            float a20 = c[IW - 1],  a21 = c[IW],  a22 = c[IW + 1];
            float corners = (a00 + a02) + (a20 + a22);
            float gxx = corners + 2.0f * (a10 + a12) - 2.0f * (a01 + a21) - 4.0f * a11;
            float gyy = corners + 2.0f * (a01 + a21) - 2.0f * (a10 + a12) - 4.0f * a11;
            float gxy = (a02 - a00) + (a20 - a22);
            float det = gxx * gyy - gxy * gxy;
            v = fmaxf(fabsf(det), 1e-6f);
        }
        s_sc[idx] = v;
    }
    __syncthreads();

    // ---- NMS on the 64x64 tile + per-block pre-NMS max ----
    // Output stores stay RT: they are re-read by the normalize pass and the
    // whole 192MB output fits the 192MB L2.
    float lmax = 1e-6f;
    const int ox = tid & 63;
    const int oy0 = tid >> 6;   // 0..3
    size_t outbase = (size_t)plane * PLANE_ELEMS + (size_t)by * IMG_W + bx;
    #pragma unroll
    for (int k = 0; k < 16; ++k) {
        int oy = oy0 + (k << 2);
        const float* c = &s_sc[(oy + 1) * SW + (ox + 1)];
        float v = c[0];
        float m = v;
        m = fmaxf(m, c[-SW - 1]); m = fmaxf(m, c[-SW]); m = fmaxf(m, c[-SW + 1]);
        m = fmaxf(m, c[-1]);                            m = fmaxf(m, c[1]);
        m = fmaxf(m, c[SW - 1]);  m = fmaxf(m, c[SW]);  m = fmaxf(m, c[SW + 1]);
        out[outbase + (size_t)oy * IMG_W + ox] = (v == m) ? v : 0.0f;
        lmax = fmaxf(lmax, v);
    }

    s_red[tid] = lmax;
    __syncthreads();
    #pragma unroll
    for (int s = 128; s > 0; s >>= 1) {
        if (tid < s) s_red[tid] = fmaxf(s_red[tid], s_red[tid + s]);
        __syncthreads();
    }
    if (tid == 0) {
        // positive floats: IEEE order == unsigned-int order
        atomicMax(&pmax[plane], __float_as_uint(s_red[0]));
    }
}

__global__ __launch_bounds__(256) void hessian_normalize(float* __restrict__ out,
                                                         const unsigned int* __restrict__ pmax) {
    size_t i = (size_t)blockIdx.x * blockDim.x + threadIdx.x;   // float4 index
    size_t elem = i << 2;
    int plane = (int)(elem >> 20);                              // 2^20 elems per plane
    float smax = __uint_as_float(pmax[plane]);
    v4f* p = reinterpret_cast<v4f*>(out) + i;
    v4f v = *p;                        // RT load: should hit in L2
    v.x /= smax; v.y /= smax; v.z /= smax; v.w /= smax;
    __builtin_nontemporal_store(v, p); // final result, never re-read
}

__global__ __launch_bounds__(64) void hessian_init_max(unsigned int* __restrict__ pmax) {
    int i = threadIdx.x;
    if (i < NPLANES) pmax[i] = __float_as_uint(1e-6f);
}

extern "C" void kernel_launch(void* const* d_in, const int* in_sizes, int n_in,
                              void* d_out, int out_size, void* d_ws, size_t ws_size,
                              hipStream_t stream) {
    (void)in_sizes; (void)n_in; (void)out_size; (void)ws_size;
    const float* x = (const float*)d_in[0];
    float* out = (float*)d_out;
    unsigned int* pmax = (unsigned int*)d_ws;   // 48 * 4 bytes of scratch

    hessian_init_max<<<1, 64, 0, stream>>>(pmax);

    dim3 grid(IMG_W / TILE, IMG_H / TILE, NPLANES);   // 16 x 16 x 48
    hessian_nms_kernel<<<grid, 256, 0, stream>>>(x, out, pmax);

    size_t nvec4 = ((size_t)NPLANES * PLANE_ELEMS) / 4;   // 12,582,912
    hessian_normalize<<<(unsigned)(nvec4 / 256), 256, 0, stream>>>(out, pmax);
}